// RISVQCLayer_75883482185872
// MI455X (gfx1250) — compile-verified
//
#include <hip/hip_runtime.h>
#include <math.h>

typedef float v2f __attribute__((ext_vector_type(2)));
typedef float v8f __attribute__((ext_vector_type(8)));

#define HID 256
#define NCOL 64
#define ROWS_PER_WG 128
#define ANG_STRIDE 68   // padded row stride (floats) to avoid LDS bank conflicts

// Branchless fast tanh: 1 - 2/(exp(2x)+1).
// exp(2x)->inf gives 1.0, exp(2x)->0 gives -1.0; no divergent slow path.
__device__ __forceinline__ float fast_tanhf(float x) {
    float e = __expf(2.0f * x);
    return 1.0f - __fdividef(2.0f, e + 1.0f);
}

__device__ __forceinline__ void ry_gate(float st[16], int w, float c, float s) {
    const int mask = 1 << (3 - w);
#pragma unroll
    for (int i = 0; i < 16; ++i) {
        if ((i & mask) == 0) {
            float s0 = st[i];
            float s1 = st[i | mask];
            st[i]        = c * s0 - s * s1;
            st[i | mask] = s * s0 + c * s1;
        }
    }
}

__device__ __forceinline__ void cnot_gate(float st[16], int cw, int tw) {
    const int cm = 1 << (3 - cw);
    const int tm = 1 << (3 - tw);
#pragma unroll
    for (int i = 0; i < 16; ++i) {
        if ((i & cm) != 0 && (i & tm) == 0) {
            float tmp = st[i];
            st[i] = st[i | tm];
            st[i | tm] = tmp;
        }
    }
}

__global__ __launch_bounds__(256) void risvqc_fused_kernel(
    const float* __restrict__ fused,   // (B, 256)
    const float* __restrict__ W,       // (64, 256)
    const float* __restrict__ bias,    // (64,)
    const float* __restrict__ vqcw,    // (16, 4, 4) flat = 256
    float* __restrict__ out)           // (B, 64)
{
    __shared__ __align__(16) float ang[ROWS_PER_WG * ANG_STRIDE];
    __shared__ float csC[256];
    __shared__ float csS[256];

    const int tid = threadIdx.x;
    const float PI_F = 3.14159265358979323846f;

    // ---- phase 0: precompute cos/sin of variational weight angles (batch invariant)
    {
        float th = 0.5f * vqcw[tid];     // index tid == r*16 + layer*4 + wire
        csC[tid] = __cosf(th);
        csS[tid] = __sinf(th);
    }

    // ---- phase 1: angles = tanh(fused @ W^T + b) * pi, via fp32 WMMA 16x16x4
    const int wave = tid >> 5;
    const int lane = tid & 31;
    const int half = lane >> 4;     // 0: K={0,1}, 1: K={2,3} within k-block
    const int l16  = lane & 15;
    const int m0   = blockIdx.x * ROWS_PER_WG + wave * 16;

    // A fragment: lane row = m0+l16, float2 at k = 4*kb + 2*half
    const float* aptr = fused + (size_t)(m0 + l16) * HID + 2 * half;
    // B fragment: B[k][n] = W[n][k]; lane col = nt*16 + l16, float2 at k = 4*kb + 2*half
    const float* bptr = W + (size_t)l16 * HID + 2 * half;

    v8f acc[4] = {};   // 4 N-tiles of 16 columns each

#pragma unroll 4
    for (int kb = 0; kb < HID / 4; ++kb) {
        v2f a = *(const v2f*)(aptr + 4 * kb);
#pragma unroll
        for (int nt = 0; nt < 4; ++nt) {
            v2f b = *(const v2f*)(bptr + nt * 16 * HID + 4 * kb);
            acc[nt] = __builtin_amdgcn_wmma_f32_16x16x4_f32(
                /*neg_a=*/false, a, /*neg_b=*/false, b,
                /*c_mod=*/(short)0, acc[nt],
                /*reuse_a=*/false, /*reuse_b=*/false);
        }
    }

    // C/D layout: lane holds column n = nt*16+l16, rows m = 8*half + j (j = VGPR idx)
#pragma unroll
    for (int nt = 0; nt < 4; ++nt) {
        int n = nt * 16 + l16;
        float bn = bias[n];
#pragma unroll
        for (int j = 0; j < 8; ++j) {
            int ml = wave * 16 + half * 8 + j;
            ang[ml * ANG_STRIDE + n] = PI_F * fast_tanhf(acc[nt][j] + bn);
        }
    }

    __syncthreads();

    // ---- phase 2: 4-qubit statevector sim per (b, r). r is fixed per thread.
    const int r = tid & 15;
    float wc[16], wsn[16];
#pragma unroll
    for (int g = 0; g < 16; ++g) {
        wc[g]  = csC[r * 16 + g];
        wsn[g] = csS[r * 16 + g];
    }

#pragma unroll 1
    for (int it = 0; it < 8; ++it) {
        int bl = (tid >> 4) + it * 16;   // local batch row 0..127
        const float4 a4 = *(const float4*)(&ang[bl * ANG_STRIDE + 4 * r]);

        float st[16];
#pragma unroll
        for (int i = 0; i < 16; ++i) st[i] = 0.0f;
        st[0] = 1.0f;

        float c, s;
        // input encoding layer
        __sincosf(0.5f * a4.x, &s, &c); ry_gate(st, 0, c, s);
        __sincosf(0.5f * a4.y, &s, &c); ry_gate(st, 1, c, s);
        __sincosf(0.5f * a4.z, &s, &c); ry_gate(st, 2, c, s);
        __sincosf(0.5f * a4.w, &s, &c); ry_gate(st, 3, c, s);
        // layer 0
        ry_gate(st, 0, wc[0], wsn[0]);  ry_gate(st, 1, wc[1], wsn[1]);
        ry_gate(st, 2, wc[2], wsn[2]);  ry_gate(st, 3, wc[3], wsn[3]);
        cnot_gate(st, 0, 1); cnot_gate(st, 2, 3);
        // layer 1
        ry_gate(st, 0, wc[4], wsn[4]);  ry_gate(st, 1, wc[5], wsn[5]);
        ry_gate(st, 2, wc[6], wsn[6]);  ry_gate(st, 3, wc[7], wsn[7]);
        cnot_gate(st, 1, 2); cnot_gate(st, 3, 0);
        // layer 2
        ry_gate(st, 0, wc[8], wsn[8]);  ry_gate(st, 1, wc[9], wsn[9]);
        ry_gate(st, 2, wc[10], wsn[10]); ry_gate(st, 3, wc[11], wsn[11]);
        cnot_gate(st, 0, 1); cnot_gate(st, 1, 2); cnot_gate(st, 2, 3); cnot_gate(st, 3, 0);
        // layer 3
        ry_gate(st, 0, wc[12], wsn[12]); ry_gate(st, 1, wc[13], wsn[13]);
        ry_gate(st, 2, wc[14], wsn[14]); ry_gate(st, 3, wc[15], wsn[15]);

        // <Z_w> = sum p[bit_w=0] - sum p[bit_w=1]
        float ev[4];
#pragma unroll
        for (int w = 0; w < 4; ++w) {
            int mask = 1 << (3 - w);
            float e = 0.0f;
#pragma unroll
            for (int i = 0; i < 16; ++i) {
                float p = st[i] * st[i];
                e += (i & mask) ? -p : p;
            }
            ev[w] = e;
        }

        float4 o;
        o.x = (ev[0] + 1.0f) * PI_F;
        o.y = (ev[1] + 1.0f) * PI_F;
        o.z = (ev[2] + 1.0f) * PI_F;
        o.w = (ev[3] + 1.0f) * PI_F;
        size_t orow = ((size_t)blockIdx.x * ROWS_PER_WG + bl) * NCOL + 4 * r;
        *(float4*)(out + orow) = o;
    }
}

extern "C" void kernel_launch(void* const* d_in, const int* in_sizes, int n_in,
                              void* d_out, int out_size, void* d_ws, size_t ws_size,
                              hipStream_t stream) {
    (void)n_in; (void)out_size; (void)d_ws; (void)ws_size;
    const float* fused = (const float*)d_in[0];   // (B, 256) fp32
    const float* Wp    = (const float*)d_in[1];   // (64, 256) fp32
    const float* bias  = (const float*)d_in[2];   // (64,) fp32
    const float* vqcw  = (const float*)d_in[3];   // (16,4,4) fp32
    float* out = (float*)d_out;                   // (B, 64) fp32

    const int Bn = in_sizes[0] / HID;             // 65536
    dim3 grid(Bn / ROWS_PER_WG);                  // 512
    dim3 block(256);                              // 8 wave32
    risvqc_fused_kernel<<<grid, block, 0, stream>>>(fused, Wp, bias, vqcw, out);
}